// TriangleMixMaskGenerator_82179904242118
// MI455X (gfx1250) — compile-verified
//
#include <hip/hip_runtime.h>
#include <math.h>

// TriangleMix mask generator for MI455X (gfx1250).
//
// Pure store-bandwidth problem: N*N f32 written (268 MB @ N=8192), nothing
// read but one uniform scalar (use_tri). Mask predicate folds to:
//   allowed = (i>=j) && (!use_tri || j<=SINK || (i-j)<=WIN || (N-i)<LAST)
//
// Issue-roofline tuning: 4x b128 NT stores per thread (64 B/lane, 2 KB/wave)
// so the per-wave SALU preamble + VALU predicate work is ~17 instructions per
// 512 B stored -- ~3x front-end headroom against the 23.3 TB/s HBM ceiling.

#define SINK_TOKENS   4
#define WINDOW_SIZE   32
#define LAST_TOKENS   64
#define CHUNKS_PER_THREAD 4   // 4 x float4 = 64 B per lane

typedef float v4f __attribute__((ext_vector_type(4)));

__global__ void __launch_bounds__(256)
tri_mask_v4x4(const int* __restrict__ use_tri_p, float* __restrict__ out, int N) {
    const int i = blockIdx.y;                    // query row (uniform)
    // chunk index base; chunks strided by blockDim for coalescing
    const int c0 = blockIdx.x * (256 * CHUNKS_PER_THREAD) + threadIdx.x;

    const int  use_tri = use_tri_p[0];           // uniform s_load
    // Rows where the mask degenerates to plain causal (uniform per block):
    const bool relax = (use_tri == 0) || ((N - i) < LAST_TOKENS);

    const float NEG = -__builtin_inff();
    float* row = out + (size_t)i * (size_t)N;

#pragma unroll
    for (int c = 0; c < CHUNKS_PER_THREAD; ++c) {
        const int j0 = (c0 + c * 256) * 4;       // key col base for this chunk
        if (j0 < N) {
            v4f v;
#pragma unroll
            for (int k = 0; k < 4; ++k) {
                const int  j       = j0 + k;
                const bool allowed = (i >= j) &&
                                     (relax || (j <= SINK_TOKENS) ||
                                      ((i - j) <= WINDOW_SIZE));
                v[k] = allowed ? 0.0f : NEG;
            }
            __builtin_nontemporal_store(v, (v4f*)(row + j0));  // b128, TH=NT
        }
    }
}

// Scalar fallback for N not divisible by 4 (keeps correctness general).
__global__ void __launch_bounds__(256)
tri_mask_s(const int* __restrict__ use_tri_p, float* __restrict__ out, int N) {
    const int i = blockIdx.y;
    const int j = blockIdx.x * 256 + threadIdx.x;
    if (j >= N) return;

    const int  use_tri = use_tri_p[0];
    const bool relax   = (use_tri == 0) || ((N - i) < LAST_TOKENS);
    const bool allowed = (i >= j) &&
                         (relax || (j <= SINK_TOKENS) || ((i - j) <= WINDOW_SIZE));
    out[(size_t)i * (size_t)N + (size_t)j] = allowed ? 0.0f : -__builtin_inff();
}

extern "C" void kernel_launch(void* const* d_in, const int* in_sizes, int n_in,
                              void* d_out, int out_size, void* d_ws, size_t ws_size,
                              hipStream_t stream) {
    (void)in_sizes; (void)n_in; (void)d_ws; (void)ws_size;

    // Inputs (setup_inputs order): d_in[0] = seq_len (int), d_in[1] = use_triangle (int).
    // Scalars live in device memory; derive N host-side from out_size = N*N.
    const int* use_tri = (const int*)d_in[1];
    float*     out     = (float*)d_out;

    const int N = (int)(sqrt((double)out_size) + 0.5);
    if (N <= 0) return;

    if ((N & 3) == 0) {
        const int chunksPerRow  = N >> 2;                       // float4 chunks per row
        const int chunksPerBlk  = 256 * CHUNKS_PER_THREAD;      // 1024 chunks = 4096 floats
        dim3 grid((chunksPerRow + chunksPerBlk - 1) / chunksPerBlk, N);
        tri_mask_v4x4<<<grid, dim3(256), 0, stream>>>(use_tri, out, N);
    } else {
        dim3 grid((N + 255) / 256, N);
        tri_mask_s<<<grid, 256, 0, stream>>>(use_tri, out, N);
    }
}